// RegionProposalNetwork_70540542869594
// MI455X (gfx1250) — compile-verified
//
#include <hip/hip_runtime.h>
#include <hip/hip_bf16.h>
#include <stdint.h>

typedef __attribute__((ext_vector_type(2))) float v2f;
typedef __attribute__((ext_vector_type(8))) float v8f;

#define N_ANCH   159882
#define PRE_NMS  4000
#define POST_NMS 1000
#define IOU_TH   0.7f
#define MIN_SZ   0.01f

// CDNA5 async global->LDS path (ASYNCcnt), guarded so either toolchain compiles.
#if defined(__has_builtin)
#if __has_builtin(__builtin_amdgcn_global_load_async_to_lds_b128) && \
    __has_builtin(__builtin_amdgcn_s_wait_asynccnt)
#define ASYNC_LDS 1
#endif
#endif

#ifdef ASYNC_LDS
typedef int v4i_vs __attribute__((vector_size(4 * sizeof(int))));
typedef __attribute__((address_space(1))) v4i_vs g_v4i;   // global (__device__)
typedef __attribute__((address_space(3))) v4i_vs l_v4i;   // LDS (__shared__)
__device__ __forceinline__ void async_b128(const float* g, float* l) {
  __builtin_amdgcn_global_load_async_to_lds_b128((g_v4i*)g, (l_v4i*)l, 0, 0);
}
#endif

// ---------------------------------------------------------------- anchors
__global__ __launch_bounds__(256) void anchors_kernel(float* __restrict__ anc) {
  int i = blockIdx.x * 256 + threadIdx.x;
  if (i >= N_ANCH) return;
  const int   bases[6]   = {0, 120000, 150000, 157500, 159375, 159882};
  const int   gws[5]     = {200, 100, 50, 25, 13};
  const float sizes[5]   = {32.f, 64.f, 128.f, 256.f, 512.f};
  const float strides[5] = {4.f, 8.f, 16.f, 32.f, 61.f};
  int l = 0;
  while (i >= bases[l + 1]) ++l;
  int rem = i - bases[l];
  int cell = rem / 3;
  int a = rem - cell * 3;
  int gw = gws[l];
  int y = cell / gw;
  int x = cell - y * gw;
  const float hr_tab[3] = {0.70710678f, 1.0f, 1.41421356f};
  float hr  = hr_tab[a];
  float wsz = sizes[l] / hr;       // s * (1/sqrt(aspect))
  float hsz = sizes[l] * hr;       // s * sqrt(aspect)
  float bx0 = rintf(-wsz * 0.5f), by0 = rintf(-hsz * 0.5f);
  float bx1 = rintf( wsz * 0.5f), by1 = rintf( hsz * 0.5f);
  float sx = (float)x * strides[l], sy = (float)y * strides[l];
  const float inv = 1.0f / 800.0f;
  anc[i * 4 + 0] = (sx + bx0) * inv;
  anc[i * 4 + 1] = (sy + by0) * inv;
  anc[i * 4 + 2] = (sx + bx1) * inv;
  anc[i * 4 + 3] = (sy + by1) * inv;
}

// --------------------------------------------------- weight repack to k-major
// wT[k][oc], k = tap*256 + ic  (tap = ty*3+tx)
__global__ __launch_bounds__(256) void transpose_w(const float* __restrict__ w,
                                                   float* __restrict__ wT) {
  int i = blockIdx.x * 256 + threadIdx.x;     // 2304*256 elements
  if (i >= 2304 * 256) return;
  int k  = i >> 8;
  int oc = i & 255;
  int tap = k / 256;
  int ic  = k - tap * 256;
  wT[k * 256 + oc] = w[(oc * 256 + ic) * 9 + tap];
}

// ------------------------------------------------------- conv3x3 + ReLU, WMMA
// C[M=gh*gw, N=256] = A[M, 2304] x B[2304, 256];  A = im2col(feat) on the fly.
// Block: 256 threads = 8 wave32, tile 32 pos x 64 oc; wave = (posHalf, ocQuad).
#define TILE_M 32
#define TILE_N 64
#define KCH    64
#define AP     40    // LDS row stride (pos) : 2*40 % 64 == 16 -> conflict-free halves
#define BP     72    // LDS row stride (oc)  : 2*72 % 64 == 16 ; 72*4 % 16 == 0 (b128 ok)

__global__ __launch_bounds__(256) void conv3x3_wmma(
    const float* __restrict__ feat,   // [256, gh, gw]
    const float* __restrict__ wT,     // [2304, 256]
    const float* __restrict__ bias,   // [256]
    float* __restrict__ h,            // [M, 256]  (NHWC)
    int gh, int gw)
{
  const int M = gh * gw;
  const int tileBase = blockIdx.x * TILE_M;
  const int ocBase   = blockIdx.y * TILE_N;
  const int tid  = threadIdx.x;
  const int wave = tid >> 5;
  const int lane = tid & 31;
  const int posHalf = wave & 1;
  const int ocQuad  = wave >> 1;

  __shared__ float As[KCH * AP];
  __shared__ float Bs[KCH * BP];

  v8f acc = {};
  const int HW = M;

  for (int kc = 0; kc < 2304; kc += KCH) {
    if (kc + KCH < 2304)
      __builtin_prefetch(&wT[(size_t)(kc + KCH) * 256 + ocBase], 0, 1);

#ifdef ASYNC_LDS
    // stage B asynchronously: memory -> LDS, no VGPR round-trip (ASYNCcnt)
    #pragma unroll
    for (int i = 0; i < (KCH * TILE_N) / (256 * 4); ++i) {   // 4 iters x 16B/lane
      int base = tid * 4 + i * 1024;
      int ocl  = base & 63;
      int kl   = base >> 6;
      async_b128(&wT[(size_t)(kc + kl) * 256 + ocBase + ocl], &Bs[kl * BP + ocl]);
    }
#else
    #pragma unroll
    for (int i = 0; i < (KCH * TILE_N) / 256; ++i) {         // 16 iters
      int idx = tid + i * 256;
      int ocl = idx & 63;
      int kl  = idx >> 6;
      Bs[kl * BP + ocl] = wT[(size_t)(kc + kl) * 256 + ocBase + ocl];
    }
#endif

    // stage A: KCH k-rows x 32 positions (bounds-checked im2col gather)
    #pragma unroll
    for (int i = 0; i < (KCH * TILE_M) / 256; ++i) {       // 8 iters
      int idx = tid + i * 256;
      int kl  = idx >> 5;
      int p   = idx & 31;
      int kg  = kc + kl;
      int tap = kg / 256;
      int ic  = kg - tap * 256;
      int pos = tileBase + p;
      float v = 0.f;
      if (pos < M) {
        int y  = pos / gw, x = pos - y * gw;
        int yy = y + tap / 3 - 1;
        int xx = x + (tap % 3) - 1;
        if (yy >= 0 && yy < gh && xx >= 0 && xx < gw)
          v = feat[(size_t)ic * HW + yy * gw + xx];
      }
      As[kl * AP + p] = v;
    }

#ifdef ASYNC_LDS
    __builtin_amdgcn_s_wait_asynccnt(0);
#endif
    __syncthreads();

    // 16 x V_WMMA_F32_16X16X4_F32 per chunk
    const int mn   = lane & 15;          // M row (A) / N col (B)
    const int koff = (lane >> 4) * 2;    // lanes16-31 hold K=2,3
    #pragma unroll
    for (int ks = 0; ks < KCH; ks += 4) {
      v2f a, b;
      a.x = As[(ks + koff + 0) * AP + posHalf * 16 + mn];
      a.y = As[(ks + koff + 1) * AP + posHalf * 16 + mn];
      b.x = Bs[(ks + koff + 0) * BP + ocQuad * 16 + mn];
      b.y = Bs[(ks + koff + 1) * BP + ocQuad * 16 + mn];
      acc = __builtin_amdgcn_wmma_f32_16x16x4_f32(
          /*neg_a=*/false, a, /*neg_b=*/false, b,
          /*c_mod=*/(short)0, acc, /*reuse_a=*/false, /*reuse_b=*/false);
    }
    __syncthreads();
  }

  // epilogue: C layout -> VGPR r, lanes0-15: M=r, lanes16-31: M=r+8; N=lane&15
  const int n    = lane & 15;
  const int mAdd = (lane >> 4) * 8;
  int oc   = ocBase + ocQuad * 16 + n;
  float bv = bias[oc];
  #pragma unroll
  for (int r = 0; r < 8; ++r) {
    int pos = tileBase + posHalf * 16 + r + mAdd;
    if (pos < M) {
      float v = acc[r] + bv;
      h[(size_t)pos * 256 + oc] = v > 0.f ? v : 0.f;
    }
  }
}

// ----------------------------------------- 1x1 heads + softmax + box decode
__global__ __launch_bounds__(256) void head_decode_kernel(
    const float* __restrict__ h,
    const float* __restrict__ w_cls, const float* __restrict__ b_cls,
    const float* __restrict__ w_reg, const float* __restrict__ b_reg,
    const float* __restrict__ anc,
    float* __restrict__ cls_out, float* __restrict__ reg_out,
    float* __restrict__ scores, float* __restrict__ boxes,
    int M, int anchorBase)
{
  __shared__ float wsh[18 * 256];
  __shared__ float bsh[18];
  int t = threadIdx.x;
#ifdef ASYNC_LDS
  #pragma unroll
  for (int i = 0; i < 5; ++i) {                  // ceil(4608 / (256*4))
    int base = t * 4 + i * 1024;
    if (base < 18 * 256) {
      int o = base >> 8, c = base & 255;
      const float* src = (o < 6) ? &w_cls[o * 256 + c] : &w_reg[(o - 6) * 256 + c];
      async_b128(src, &wsh[base]);
    }
  }
#else
  for (int i = t; i < 18 * 256; i += 256) {
    int o = i >> 8, c = i & 255;
    wsh[i] = (o < 6) ? w_cls[o * 256 + c] : w_reg[(o - 6) * 256 + c];
  }
#endif
  if (t < 18) bsh[t] = (t < 6) ? b_cls[t] : b_reg[t - 6];
#ifdef ASYNC_LDS
  __builtin_amdgcn_s_wait_asynccnt(0);
#endif
  __syncthreads();

  int p = blockIdx.x * 256 + t;
  if (p >= M) return;

  float acc[18];
  #pragma unroll
  for (int o = 0; o < 18; ++o) acc[o] = bsh[o];
  const float* hp = h + (size_t)p * 256;
  for (int c = 0; c < 256; ++c) {
    float hv = hp[c];
    #pragma unroll
    for (int o = 0; o < 18; ++o) acc[o] = fmaf(hv, wsh[o * 256 + c], acc[o]);
  }

  #pragma unroll
  for (int a = 0; a < 3; ++a) {
    int idx = anchorBase + p * 3 + a;
    float c0 = acc[a * 2 + 0], c1 = acc[a * 2 + 1];
    cls_out[(size_t)idx * 2 + 0] = c0;
    cls_out[(size_t)idx * 2 + 1] = c1;
    float t0 = acc[6 + a * 4 + 0], t1 = acc[6 + a * 4 + 1];
    float t2 = acc[6 + a * 4 + 2], t3 = acc[6 + a * 4 + 3];
    reg_out[(size_t)idx * 4 + 0] = t0;
    reg_out[(size_t)idx * 4 + 1] = t1;
    reg_out[(size_t)idx * 4 + 2] = t2;
    reg_out[(size_t)idx * 4 + 3] = t3;

    float a0 = anc[(size_t)idx * 4 + 0], a1 = anc[(size_t)idx * 4 + 1];
    float a2 = anc[(size_t)idx * 4 + 2], a3 = anc[(size_t)idx * 4 + 3];
    float aw = a2 - a0, ah = a3 - a1;
    float acx = 0.5f * (a0 + a2), acy = 0.5f * (a1 + a3);
    float cx = t0 * aw + acx, cy = t1 * ah + acy;
    float bw = expf(t2) * aw,  bh = expf(t3) * ah;
    float x0 = fminf(fmaxf(cx - 0.5f * bw, 0.f), 1.f);
    float y0 = fminf(fmaxf(cy - 0.5f * bh, 0.f), 1.f);
    float x1 = fminf(fmaxf(cx + 0.5f * bw, 0.f), 1.f);
    float y1 = fminf(fmaxf(cy + 0.5f * bh, 0.f), 1.f);
    bool valid = ((x1 - x0) >= MIN_SZ) && ((y1 - y0) >= MIN_SZ);
    float mm = fmaxf(c0, c1);
    float e0 = expf(c0 - mm), e1 = expf(c1 - mm);
    float s  = e1 / (e0 + e1);
    scores[idx] = valid ? s : -INFINITY;
    boxes[(size_t)idx * 4 + 0] = x0;
    boxes[(size_t)idx * 4 + 1] = y0;
    boxes[(size_t)idx * 4 + 2] = x1;
    boxes[(size_t)idx * 4 + 3] = y1;
  }
}

// ----------------------------------------------------------- top-k machinery
// state: st[0..255] hist, st[256] prefix, st[257] kRem, st[258] candCount
__device__ __forceinline__ uint32_t score_key(float s) {
  uint32_t u = __float_as_uint(s);
  return (u & 0x80000000u) ? ~u : (u | 0x80000000u);
}

__global__ void init_state(uint32_t* st) {
  int t = threadIdx.x;
  st[t] = 0;
  if (t == 0) { st[256] = 0; st[257] = PRE_NMS; st[258] = 0; }
}

__global__ __launch_bounds__(256) void hist_pass(const float* __restrict__ scores,
                                                 uint32_t* st, int N, int pass) {
  __shared__ uint32_t lh[256];
  int t = threadIdx.x;
  lh[t] = 0;
  __syncthreads();
  int i = blockIdx.x * 256 + t;
  if (i < N) {
    uint32_t key = score_key(scores[i]);
    uint32_t maskHi = (pass == 3) ? 0u : (0xFFFFFFFFu << ((pass + 1) * 8));
    if ((key & maskHi) == st[256]) {
      uint32_t d = (key >> (pass * 8)) & 0xFFu;
      atomicAdd(&lh[d], 1u);
    }
  }
  __syncthreads();
  if (lh[t]) atomicAdd(&st[t], lh[t]);
}

__global__ __launch_bounds__(256) void select_pass(uint32_t* st, int pass) {
  if (threadIdx.x == 0) {
    uint32_t k = st[257];
    uint32_t cum = 0;
    int b = 0;
    for (int d = 255; d >= 0; --d) {
      uint32_t c = st[d];
      if (cum + c >= k) { b = d; break; }
      cum += c;
    }
    st[256] |= ((uint32_t)b) << (pass * 8);
    st[257] = k - cum;
  }
  __syncthreads();
  st[threadIdx.x] = 0;   // zero hist for next pass
}

__global__ __launch_bounds__(256) void collect_cand(const float* __restrict__ scores,
                                                    uint32_t* st,
                                                    uint32_t* __restrict__ candKey,
                                                    uint32_t* __restrict__ candIdx,
                                                    int N) {
  int i = blockIdx.x * 256 + threadIdx.x;
  if (i >= N) return;
  uint32_t key = score_key(scores[i]);
  if (key >= st[256]) {
    uint32_t j = atomicAdd(&st[258], 1u);
    if (j < 4096u) { candKey[j] = key; candIdx[j] = (uint32_t)i; }
  }
}

// one-block bitonic sort of 4096 (key<<32 | ~idx), descending; emit ranked idx
__global__ __launch_bounds__(1024) void sort_topk(const uint32_t* __restrict__ candKey,
                                                  const uint32_t* __restrict__ candIdx,
                                                  const uint32_t* st,
                                                  uint32_t* __restrict__ rankIdx) {
  __shared__ unsigned long long sd[4096];
  int t = threadIdx.x;
  uint32_t cnt = st[258];
  if (cnt > 4096u) cnt = 4096u;
  for (int i = t; i < 4096; i += 1024) {
    unsigned long long v = 0ull;
    if (i < (int)cnt)
      v = ((unsigned long long)candKey[i] << 32) | (unsigned long long)(~candIdx[i]);
    sd[i] = v;
  }
  __syncthreads();
  for (int ksz = 2; ksz <= 4096; ksz <<= 1) {
    for (int j = ksz >> 1; j > 0; j >>= 1) {
      #pragma unroll
      for (int s = 0; s < 4; ++s) {
        int i = t + s * 1024;
        int ixj = i ^ j;
        if (ixj > i) {
          bool descBlock = ((i & ksz) == 0);
          unsigned long long vi = sd[i], vj = sd[ixj];
          bool swap = descBlock ? (vi < vj) : (vi > vj);
          if (swap) { sd[i] = vj; sd[ixj] = vi; }
        }
      }
      __syncthreads();
    }
  }
  for (int r = t; r < PRE_NMS; r += 1024) {
    unsigned long long v = sd[r];
    rankIdx[r] = (v == 0ull) ? 0xFFFFFFFFu : ~((uint32_t)(v & 0xFFFFFFFFull));
  }
}

// -------------------------------------------- gather + greedy NMS + compact
__global__ __launch_bounds__(1024) void nms_out_kernel(
    const uint32_t* __restrict__ rankIdx,
    const float* __restrict__ scores,
    const float* __restrict__ boxesAll,
    float* __restrict__ boxesTop,     // ws [PRE_NMS*4]
    float* __restrict__ roi_out)      // d_out [POST_NMS*4]
{
  __shared__ unsigned char keep[PRE_NMS];
  int t = threadIdx.x;
  for (int r = t; r < PRE_NMS; r += 1024) {
    uint32_t idx = rankIdx[r];
    float s = (idx != 0xFFFFFFFFu) ? scores[idx] : -INFINITY;
    bool fin = (idx != 0xFFFFFFFFu) && isfinite(s);
    float b0 = 0.f, b1 = 0.f, b2 = 0.f, b3 = 0.f;
    if (fin) {
      b0 = boxesAll[(size_t)idx * 4 + 0];
      b1 = boxesAll[(size_t)idx * 4 + 1];
      b2 = boxesAll[(size_t)idx * 4 + 2];
      b3 = boxesAll[(size_t)idx * 4 + 3];
    }
    boxesTop[r * 4 + 0] = b0;
    boxesTop[r * 4 + 1] = b1;
    boxesTop[r * 4 + 2] = b2;
    boxesTop[r * 4 + 3] = b3;
    keep[r] = fin ? 1 : 0;
  }
  for (int i = t; i < POST_NMS * 4; i += 1024) roi_out[i] = 0.f;
  __syncthreads();

  for (int i = 0; i < PRE_NMS; ++i) {
    if (keep[i]) {
      float bi0 = boxesTop[i * 4 + 0], bi1 = boxesTop[i * 4 + 1];
      float bi2 = boxesTop[i * 4 + 2], bi3 = boxesTop[i * 4 + 3];
      float areaI = fmaxf(bi2 - bi0, 0.f) * fmaxf(bi3 - bi1, 0.f);
      for (int j = i + 1 + t; j < PRE_NMS; j += 1024) {
        if (keep[j]) {
          float bj0 = boxesTop[j * 4 + 0], bj1 = boxesTop[j * 4 + 1];
          float bj2 = boxesTop[j * 4 + 2], bj3 = boxesTop[j * 4 + 3];
          float areaJ = fmaxf(bj2 - bj0, 0.f) * fmaxf(bj3 - bj1, 0.f);
          float iw = fmaxf(fminf(bi2, bj2) - fmaxf(bi0, bj0), 0.f);
          float ih = fmaxf(fminf(bi3, bj3) - fmaxf(bi1, bj1), 0.f);
          float inter = iw * ih;
          float uni = areaI + areaJ - inter;
          float iou = (uni > 0.f) ? inter / uni : 0.f;
          if (iou > IOU_TH) keep[j] = 0;
        }
      }
    }
    __syncthreads();
  }

  if (t == 0) {
    int cnt = 0;
    for (int r = 0; r < PRE_NMS && cnt < POST_NMS; ++r) {
      if (keep[r]) {
        roi_out[cnt * 4 + 0] = boxesTop[r * 4 + 0];
        roi_out[cnt * 4 + 1] = boxesTop[r * 4 + 1];
        roi_out[cnt * 4 + 2] = boxesTop[r * 4 + 2];
        roi_out[cnt * 4 + 3] = boxesTop[r * 4 + 3];
        ++cnt;
      }
    }
  }
}

// ------------------------------------------------------------------ launcher
extern "C" void kernel_launch(void* const* d_in, const int* in_sizes, int n_in,
                              void* d_out, int out_size, void* d_ws, size_t ws_size,
                              hipStream_t stream) {
  (void)in_sizes; (void)n_in; (void)out_size; (void)ws_size;
  const float* feats[5] = {(const float*)d_in[1], (const float*)d_in[2],
                           (const float*)d_in[3], (const float*)d_in[4],
                           (const float*)d_in[5]};
  const float* w_inter = (const float*)d_in[6];
  const float* b_inter = (const float*)d_in[7];
  const float* w_cls   = (const float*)d_in[8];
  const float* b_cls   = (const float*)d_in[9];
  const float* w_reg   = (const float*)d_in[10];
  const float* b_reg   = (const float*)d_in[11];

  float* out     = (float*)d_out;
  float* cls_out = out;                       // 159882*2
  float* reg_out = out + 319764;              // 159882*4
  float* roi_out = out + 959292;              // 1000*4
  float* anc_out = out + 963292;              // 159882*4

  // workspace layout (floats)
  float* ws       = (float*)d_ws;
  float* h        = ws;                        // 10,240,000
  float* wT       = ws + 10240000;             //    589,824
  float* scores   = ws + 10829824;             //    159,882
  float* boxesAll = ws + 10989712;             //    639,528
  float* boxesTop = ws + 11629240;             //     16,000
  uint32_t* st      = (uint32_t*)(ws + 11645248);  // 272 reserved
  uint32_t* candKey = st + 272;                    // 4096
  uint32_t* candIdx = candKey + 4096;              // 4096
  uint32_t* rankIdx = candIdx + 4096;              // 4096

  anchors_kernel<<<(N_ANCH + 255) / 256, 256, 0, stream>>>(anc_out);
  transpose_w<<<(2304 * 256 + 255) / 256, 256, 0, stream>>>(w_inter, wT);
  init_state<<<1, 256, 0, stream>>>(st);

  const int GSZ[5]   = {200, 100, 50, 25, 13};
  const int ABASE[5] = {0, 120000, 150000, 157500, 159375};
  for (int l = 0; l < 5; ++l) {
    int g = GSZ[l], M = g * g;
    dim3 grid((M + TILE_M - 1) / TILE_M, 256 / TILE_N);
    conv3x3_wmma<<<grid, 256, 0, stream>>>(feats[l], wT, b_inter, h, g, g);
    head_decode_kernel<<<(M + 255) / 256, 256, 0, stream>>>(
        h, w_cls, b_cls, w_reg, b_reg, anc_out,
        cls_out, reg_out, scores, boxesAll, M, ABASE[l]);
  }

  for (int p = 3; p >= 0; --p) {
    hist_pass<<<(N_ANCH + 255) / 256, 256, 0, stream>>>(scores, st, N_ANCH, p);
    select_pass<<<1, 256, 0, stream>>>(st, p);
  }
  collect_cand<<<(N_ANCH + 255) / 256, 256, 0, stream>>>(scores, st, candKey, candIdx, N_ANCH);
  sort_topk<<<1, 1024, 0, stream>>>(candKey, candIdx, st, rankIdx);
  nms_out_kernel<<<1, 1024, 0, stream>>>(rankIdx, scores, boxesAll, boxesTop, roi_out);
}